// GraphReader_22067541967023
// MI455X (gfx1250) — compile-verified
//
#include <hip/hip_runtime.h>
#include <hip/hip_bf16.h>
#include <math.h>
#include <stdint.h>

// Problem constants (from reference)
#define B_   32
#define N_   65536
#define E_   768
#define H_   8
#define DH_  96
#define SCALE_ 0.10206207261596575f   // 1/sqrt(96)
#define NEG_INF_ (-3.402823466e38f)
#define SPLITS_ 16

typedef float v2f __attribute__((ext_vector_type(2)));
typedef float v8f __attribute__((ext_vector_type(8)));
typedef unsigned int su4 __attribute__((ext_vector_type(4)));
typedef unsigned int su8 __attribute__((ext_vector_type(8)));

// ---------------------------------------------------------------------------
// Tensor Data Mover: DMA a 2D tile (rows x row_len f32, row-major, given row
// stride) from global memory into LDS. Descriptor per CDNA5 ISA ch.8:
//   group0: [1:0]=count=1, [63:32]=lds_addr, [120:64]=global_addr, [127:126]=2
//   group1: data_size=4B (code 2), tensor_dim0/1, tile_dim0/1, dim0_stride
// Issued via inline asm (portable across toolchains); tracked by TENSORcnt.
// ---------------------------------------------------------------------------
__device__ __forceinline__ void tdm_load_2d_f32(unsigned lds_off,
                                                const float* gsrc,
                                                unsigned row_len,      // elts
                                                unsigned rows,         // tile rows
                                                unsigned stride,       // elts
                                                unsigned tensor_rows)  // OOB dim1
{
  const unsigned long long ga = (unsigned long long)(uintptr_t)gsrc;
  su4 g0;
  g0.x = 1u;                                   // count=1 (user descriptor)
  g0.y = lds_off;                              // lds_addr (bytes)
  g0.z = (unsigned)(ga & 0xffffffffull);       // global_addr[31:0]
  g0.w = ((unsigned)(ga >> 32) & 0x01ffffffu)  // global_addr[56:32]
         | 0x80000000u;                        // type=2 ("image")
  su8 g1;
  g1.s0 = 2u << 16;                                  // data_size = 4 bytes
  g1.s1 = (row_len & 0xffffu) << 16;                 // tensor_dim0[15:0]
  g1.s2 = (row_len >> 16) | ((tensor_rows & 0xffffu) << 16); // dim0 hi | dim1 lo
  g1.s3 = (tensor_rows >> 16) | ((row_len & 0xffffu) << 16); // dim1 hi | tile_dim0
  g1.s4 = rows & 0xffffu;                            // tile_dim1 (tile_dim2=0)
  g1.s5 = stride;                                    // tensor_dim0_stride lo32
  g1.s6 = 0u;                                        // stride hi | dim1_stride lo
  g1.s7 = 0u;
  asm volatile("tensor_load_to_lds %0, %1" : : "s"(g0), "s"(g1) : "memory");
}

// ---------------------------------------------------------------------------
// WMMA GEMM: C[32,768] = A[32,768] * W[768,768]^T + bias  (fp32, 16x16x4)
// grid = 48 (one per 16-col N tile), blockDim = 64 (2 waves; wave w -> M tile w).
// The shared B tile (16 rows of W, 48KB) is DMA'd into LDS by the TDM and
// reused by both waves; fragments come from LDS (ds_load_b64).
// A-frag (16x4):  v0 = {K0 | K2}, v1 = {K1 | K3}   (lanes 0-15 | 16-31)
// B-frag (4x16):  v0 = {K0 | K2}, v1 = {K1 | K3}, N = lane%16
// C/D (16x16):    VGPR j -> M = j (+8 for lanes 16-31), N = lane%16
// ---------------------------------------------------------------------------
__global__ void gemm32_wmma(const float* __restrict__ A,
                            const float* __restrict__ W,
                            const float* __restrict__ bias,
                            float* __restrict__ C) {
  __shared__ float lds_b[16 * E_];             // 48 KB: W rows [nt*16, nt*16+16)

  const int lane = threadIdx.x & 31;
  const int wv   = threadIdx.x >> 5;           // wave id = M tile
  const int half = lane >> 4;
  const int l16  = lane & 15;
  const int mt   = wv;                         // 2 tiles of M=32
  const int nt   = blockIdx.x;                 // 48 tiles of N=768

  if (threadIdx.x == 0) {
    tdm_load_2d_f32((unsigned)(uintptr_t)lds_b,
                    W + (size_t)nt * 16 * E_,
                    E_, 16, E_, 16);
    __builtin_amdgcn_s_wait_tensorcnt(0);      // issuing wave drains TENSORcnt
  }
  __syncthreads();                             // publish LDS tile to both waves

  const float* arow = A + (size_t)(mt * 16 + l16) * E_ + 2 * half;
  const float* brow = lds_b + (size_t)l16 * E_ + 2 * half;

  v8f c = {0.f, 0.f, 0.f, 0.f, 0.f, 0.f, 0.f, 0.f};
#pragma unroll 4
  for (int k = 0; k < E_; k += 4) {
    v2f a = *(const v2f*)(arow + k);
    v2f b = *(const v2f*)(brow + k);
    c = __builtin_amdgcn_wmma_f32_16x16x4_f32(false, a, false, b,
                                              (short)0, c, false, false);
  }
  const int col = nt * 16 + l16;
  const float bv = bias[col];
#pragma unroll
  for (int j = 0; j < 8; ++j) {
    const int row = mt * 16 + j + 8 * half;
    C[(size_t)row * E_ + col] = c[j] + bv;
  }
}

// ---------------------------------------------------------------------------
// Workspace init (deterministic per call)
// ---------------------------------------------------------------------------
__global__ void init_kernel(float* ctxw, float* segmax, float* segsum, int* seg) {
  const int t = blockIdx.x * blockDim.x + threadIdx.x;
  if (t < B_ * H_ * E_) ctxw[t] = 0.f;
  if (t < B_ * H_) { segmax[t] = NEG_INF_; segsum[t] = 0.f; }
  if (t <= B_) seg[t] = -1;
}

// qk[b,h,:] = Wk_h^T Q[b,h]   (Wk = W_in rows E..2E-1)
__global__ void qk_kernel(const float* __restrict__ Qm,
                          const float* __restrict__ W_in,
                          float* __restrict__ qk) {
  const int t = blockIdx.x * blockDim.x + threadIdx.x;  // B*H*E threads
  const int e = t % E_;
  const int bh = t / E_;
  const int h = bh & (H_ - 1);
  const float* q = Qm + (size_t)(bh >> 3) * E_ + h * DH_;
  const float* w = W_in + (size_t)(E_ + h * DH_) * E_ + e;
  float acc = 0.f;
#pragma unroll 4
  for (int d = 0; d < DH_; ++d) acc += q[d] * w[(size_t)d * E_];
  qk[(size_t)bh * E_ + e] = acc;
}

// s0[b,h] = Q[b,h] . bk_h
__global__ void s0_kernel(const float* __restrict__ Qm,
                          const float* __restrict__ b_in,
                          float* __restrict__ s0) {
  const int t = threadIdx.x;           // 256 = B*H
  const int h = t & (H_ - 1);
  const int b = t >> 3;
  const float* q = Qm + (size_t)b * E_ + h * DH_;
  const float* bk = b_in + E_ + h * DH_;
  float a = 0.f;
#pragma unroll 4
  for (int d = 0; d < DH_; ++d) a += q[d] * bk[d];
  s0[t] = a;
}

// segment boundaries (batch is sorted)
__global__ void segb_kernel(const int* __restrict__ batch, int* __restrict__ seg) {
  const int n = blockIdx.x * blockDim.x + threadIdx.x;
  if (n >= N_) return;
  const int b = batch[n];
  if (n == 0) seg[b] = 0;
  else if (batch[n - 1] != b) seg[b] = n;
}

__global__ void segfix_kernel(int* seg) {
  seg[B_] = N_;
  for (int b = B_ - 1; b >= 0; --b)
    if (seg[b] < 0) seg[b] = seg[b + 1];
}

// ---------------------------------------------------------------------------
// Pass A: scores[n,h] = SCALE * (x_n . qk[batch[n],h] + s0[batch[n],h])
// One thread per node, streaming its row with b128 loads (single-use data ->
// no LDS staging); qk reads are wave-uniform broadcasts served from WGP$.
// ---------------------------------------------------------------------------
__global__ void scores_kernel(const float* __restrict__ x,
                              const int* __restrict__ batch,
                              const float* __restrict__ qk,
                              const float* __restrict__ s0,
                              float* __restrict__ scores) {
  const int n = blockIdx.x * blockDim.x + threadIdx.x;
  const int b = batch[n];
  const float4* xr = (const float4*)(x + (size_t)n * E_);
  const float* qkb = qk + (size_t)b * H_ * E_;
  float acc[H_];
#pragma unroll
  for (int h = 0; h < H_; ++h) acc[h] = 0.f;

  for (int i = 0; i < E_ / 4; ++i) {
    if ((i & 15) == 0 && n < N_ - 8)
      __builtin_prefetch((const void*)(xr + i + 64), 0, 0);  // global_prefetch_b8
    const float4 xv = xr[i];
#pragma unroll
    for (int h = 0; h < H_; ++h) {
      const float4 qv = ((const float4*)(qkb + (size_t)h * E_))[i];
      acc[h] += xv.x * qv.x + xv.y * qv.y + xv.z * qv.z + xv.w * qv.w;
    }
  }
  const float* s0b = s0 + b * H_;
#pragma unroll
  for (int h = 0; h < H_; ++h)
    scores[(size_t)n * H_ + h] = (acc[h] + s0b[h]) * SCALE_;
}

// ---------------------------------------------------------------------------
// Segment softmax stats: per-graph max then sum(exp(s-max)) for all 8 heads.
// ---------------------------------------------------------------------------
__global__ void stats_kernel(const float* __restrict__ scores,
                             const int* __restrict__ seg,
                             float* __restrict__ segmax,
                             float* __restrict__ segsum) {
  __shared__ float red[H_ * 256];
  const int b = blockIdx.x, t = threadIdx.x;
  const int lo = seg[b], hi = seg[b + 1];

  float mx[H_];
#pragma unroll
  for (int h = 0; h < H_; ++h) mx[h] = NEG_INF_;
  for (int n = lo + t; n < hi; n += 256) {
    const float* sr = scores + (size_t)n * H_;
#pragma unroll
    for (int h = 0; h < H_; ++h) { const float v = sr[h]; if (v > mx[h]) mx[h] = v; }
  }
#pragma unroll
  for (int h = 0; h < H_; ++h) red[h * 256 + t] = mx[h];
  __syncthreads();
  for (int st = 128; st > 0; st >>= 1) {
    if (t < st)
#pragma unroll
      for (int h = 0; h < H_; ++h) {
        const float o = red[h * 256 + t + st];
        if (o > red[h * 256 + t]) red[h * 256 + t] = o;
      }
    __syncthreads();
  }
#pragma unroll
  for (int h = 0; h < H_; ++h) mx[h] = red[h * 256];
  __syncthreads();

  float sm[H_];
#pragma unroll
  for (int h = 0; h < H_; ++h) sm[h] = 0.f;
  for (int n = lo + t; n < hi; n += 256) {
    const float* sr = scores + (size_t)n * H_;
#pragma unroll
    for (int h = 0; h < H_; ++h) sm[h] += expf(sr[h] - mx[h]);
  }
#pragma unroll
  for (int h = 0; h < H_; ++h) red[h * 256 + t] = sm[h];
  __syncthreads();
  for (int st = 128; st > 0; st >>= 1) {
    if (t < st)
#pragma unroll
      for (int h = 0; h < H_; ++h) red[h * 256 + t] += red[h * 256 + t + st];
    __syncthreads();
  }
  if (t == 0) {
#pragma unroll
    for (int h = 0; h < H_; ++h) {
      segmax[b * H_ + h] = mx[h];
      segsum[b * H_ + h] = red[h * 256];
    }
  }
}

// w[n,h] = exp(s-max)/denom ; attn_out[n] = mean_h w[n,h]
__global__ void weights_kernel(const float* __restrict__ scores,
                               const int* __restrict__ batch,
                               const float* __restrict__ segmax,
                               const float* __restrict__ segsum,
                               float* __restrict__ wbuf,
                               float* __restrict__ attn_out) {
  const int n = blockIdx.x * blockDim.x + threadIdx.x;
  const int b = batch[n];
  const float* mx = segmax + b * H_;
  const float* ss = segsum + b * H_;
  const float* sr = scores + (size_t)n * H_;
  float sum = 0.f;
#pragma unroll
  for (int h = 0; h < H_; ++h) {
    const float denom = (ss[h] == 0.f) ? 1.f : ss[h];
    const float w = expf(sr[h] - mx[h]) / denom;
    wbuf[(size_t)n * H_ + h] = w;
    sum += w;
  }
  attn_out[n] = sum * (1.f / H_);
}

// ---------------------------------------------------------------------------
// Pass C: ctxw[b,h,:] += sum_n w[n,h] * x_n   (second stream of node_features)
// ---------------------------------------------------------------------------
__global__ void agg_kernel(const float* __restrict__ x,
                           const float* __restrict__ wbuf,
                           const int* __restrict__ seg,
                           float* __restrict__ ctxw) {
  const int b = blockIdx.x / SPLITS_;
  const int s = blockIdx.x % SPLITS_;
  const int lo = seg[b], hi = seg[b + 1];
  const int len = hi - lo;
  const int chunk = (len + SPLITS_ - 1) / SPLITS_;
  const int n0 = lo + s * chunk;
  int n1 = n0 + chunk; if (n1 > hi) n1 = hi;
  const int t = threadIdx.x;  // 256

  float acc[3][H_];
#pragma unroll
  for (int i = 0; i < 3; ++i)
#pragma unroll
    for (int h = 0; h < H_; ++h) acc[i][h] = 0.f;

  for (int n = n0; n < n1; ++n) {
    const float4 w0 = *(const float4*)(wbuf + (size_t)n * H_);
    const float4 w1 = *(const float4*)(wbuf + (size_t)n * H_ + 4);
    const float w8[H_] = {w0.x, w0.y, w0.z, w0.w, w1.x, w1.y, w1.z, w1.w};
    const float* xrow = x + (size_t)n * E_;
    if (((n - n0) & 7) == 0 && n + 8 < N_)
      __builtin_prefetch((const void*)(xrow + 8 * E_ + t), 0, 0);
    const float xv[3] = {xrow[t], xrow[t + 256], xrow[t + 512]};
#pragma unroll
    for (int i = 0; i < 3; ++i)
#pragma unroll
      for (int h = 0; h < H_; ++h) acc[i][h] += w8[h] * xv[i];
  }
#pragma unroll
  for (int i = 0; i < 3; ++i) {
    const int e = t + i * 256;
#pragma unroll
    for (int h = 0; h < H_; ++h)
      atomicAdd(&ctxw[((size_t)b * H_ + h) * E_ + e], acc[i][h]);
  }
}

// ctx[b, h*Dh+d] = Wv[h*Dh+d,:] . ctxw[b,h,:] + bv[h*Dh+d] * (nonempty)
__global__ void ctx_kernel(const float* __restrict__ W_in,
                           const float* __restrict__ b_in,
                           const float* __restrict__ ctxw,
                           const int* __restrict__ seg,
                           float* __restrict__ ctx) {
  const int t = blockIdx.x * blockDim.x + threadIdx.x;  // B*E
  const int b = t / E_;
  const int ep = t % E_;
  const int h = ep / DH_;
  const float4* wr = (const float4*)(W_in + (size_t)(2 * E_ + ep) * E_);
  const float4* sr = (const float4*)(ctxw + ((size_t)b * H_ + h) * E_);
  float a = 0.f;
  for (int i = 0; i < E_ / 4; ++i) {
    const float4 w = wr[i];
    const float4 sv = sr[i];
    a += w.x * sv.x + w.y * sv.y + w.z * sv.z + w.w * sv.w;
  }
  const float nonempty = (seg[b + 1] > seg[b]) ? 1.f : 0.f;
  ctx[t] = a + b_in[2 * E_ + ep] * nonempty;
}

// LayerNorm over E, one workgroup per row
__global__ void ln_kernel(const float* __restrict__ r,
                          const float* __restrict__ gamma,
                          const float* __restrict__ beta,
                          float* __restrict__ out) {
  __shared__ float red[256];
  const int b = blockIdx.x, t = threadIdx.x;
  const float* row = r + (size_t)b * E_;
  const float v0 = row[t], v1 = row[t + 256], v2 = row[t + 512];

  red[t] = v0 + v1 + v2;
  __syncthreads();
  for (int st = 128; st > 0; st >>= 1) {
    if (t < st) red[t] += red[t + st];
    __syncthreads();
  }
  const float mu = red[0] * (1.f / E_);
  __syncthreads();

  const float d0 = v0 - mu, d1 = v1 - mu, d2 = v2 - mu;
  red[t] = d0 * d0 + d1 * d1 + d2 * d2;
  __syncthreads();
  for (int st = 128; st > 0; st >>= 1) {
    if (t < st) red[t] += red[t + st];
    __syncthreads();
  }
  const float inv = rsqrtf(red[0] * (1.f / E_) + 1e-5f);
  out[(size_t)b * E_ + t]       = d0 * inv * gamma[t]       + beta[t];
  out[(size_t)b * E_ + t + 256] = d1 * inv * gamma[t + 256] + beta[t + 256];
  out[(size_t)b * E_ + t + 512] = d2 * inv * gamma[t + 512] + beta[t + 512];
}

// ---------------------------------------------------------------------------
extern "C" void kernel_launch(void* const* d_in, const int* in_sizes, int n_in,
                              void* d_out, int out_size, void* d_ws, size_t ws_size,
                              hipStream_t stream) {
  const float* x      = (const float*)d_in[0];   // node_features [N,E]
  const float* query  = (const float*)d_in[1];   // [B,E]
  const int*   batch  = (const int*)d_in[2];     // [N]
  const float* W_qp   = (const float*)d_in[3];
  const float* b_qp   = (const float*)d_in[4];
  const float* W_in   = (const float*)d_in[5];   // [3E,E]
  const float* b_in   = (const float*)d_in[6];   // [3E]
  const float* W_o    = (const float*)d_in[7];
  const float* b_o    = (const float*)d_in[8];
  const float* W_out  = (const float*)d_in[9];
  const float* b_out  = (const float*)d_in[10];
  const float* gamma  = (const float*)d_in[11];
  const float* beta   = (const float*)d_in[12];

  float* out = (float*)d_out;                    // [B*E] read, then [N] attn
  float* ws  = (float*)d_ws;

  // workspace layout (floats)
  float* qp     = ws + 0;              // 24576
  float* Qm     = ws + 24576;          // 24576
  float* qk     = ws + 49152;          // 196608
  float* s0     = ws + 245760;         // 256
  float* segmax = ws + 246016;         // 256
  float* segsum = ws + 246272;         // 256
  int*   seg    = (int*)(ws + 246528); // 33 ints (padded to 256)
  float* scores = ws + 246784;         // 524288
  float* wbuf   = ws + 771072;         // 524288
  float* ctxw   = ws + 1295360;        // 196608
  float* ctx    = ws + 1491968;        // 24576
  float* mha    = ws + 1516544;        // 24576
  float* read0  = ws + 1541120;        // 24576

  init_kernel<<<768, 256, 0, stream>>>(ctxw, segmax, segsum, seg);

  // projector chain: qp = query@Wqp^T+b ; Q = qp@Wq^T+bq
  gemm32_wmma<<<48, 64, 0, stream>>>(query, W_qp, b_qp, qp);
  gemm32_wmma<<<48, 64, 0, stream>>>(qp, W_in, b_in, Qm);   // Wq = rows 0..E-1

  qk_kernel<<<768, 256, 0, stream>>>(Qm, W_in, qk);
  s0_kernel<<<1, 256, 0, stream>>>(Qm, b_in, s0);

  segb_kernel<<<N_ / 256, 256, 0, stream>>>(batch, seg);
  segfix_kernel<<<1, 1, 0, stream>>>(seg);

  scores_kernel<<<N_ / 256, 256, 0, stream>>>(x, batch, qk, s0, scores);
  stats_kernel<<<B_, 256, 0, stream>>>(scores, seg, segmax, segsum);
  weights_kernel<<<N_ / 256, 256, 0, stream>>>(scores, batch, segmax, segsum,
                                               wbuf, out + B_ * E_);
  agg_kernel<<<B_ * SPLITS_, 256, 0, stream>>>(x, wbuf, seg, ctxw);

  ctx_kernel<<<(B_ * E_) / 256, 256, 0, stream>>>(W_in, b_in, ctxw, seg, ctx);
  gemm32_wmma<<<48, 64, 0, stream>>>(ctx, W_o, b_o, mha);
  gemm32_wmma<<<48, 64, 0, stream>>>(mha, W_out, b_out, read0);
  ln_kernel<<<B_, 256, 0, stream>>>(read0, gamma, beta, out);
}